// WeightedProcrustes_23914377904896
// MI455X (gfx1250) — compile-verified
//
#include <hip/hip_runtime.h>
#include <hip/hip_bf16.h>

// CDNA5 / gfx1250. wave32. WMMA f32 16x16x4 used as a K=4 f32 reduction engine.

typedef __attribute__((ext_vector_type(2))) float v2f;
typedef __attribute__((ext_vector_type(8))) float v8f;

#define WP_THRESH 0.5f
#define WP_EPS    1e-5f

static constexpr int NPTS  = 2048;   // points per batch (reference shape)
static constexpr int BLKT  = 256;    // 8 waves of 32
static constexpr int WAVES = BLKT / 32;
static constexpr int PPW   = NPTS / WAVES;   // 256 points per wave

// --------------------------------------------------------------------------
// Per-batch kernel: stream points, accumulate 4x4 Gram G = sum a b^T via
// V_WMMA_F32_16X16X4_F32 (4 groups of 4 points per instruction), reduce in
// LDS, then thread 0 solves the 3x3 Kabsch problem (Jacobi eigensolver).
// --------------------------------------------------------------------------
__global__ __launch_bounds__(BLKT)
void wproc_kernel(const float* __restrict__ src,
                  const float* __restrict__ tgt,
                  const float* __restrict__ wts,
                  float* __restrict__ out, int B)
{
    const int b    = blockIdx.x;
    const int tid  = threadIdx.x;
    const int lane = tid & 31;
    const int wave = tid >> 5;

    __shared__ float Hl[16];           // G[d][e] at Hl[d*4+e]
    if (tid < 16) Hl[tid] = 0.0f;
    __syncthreads();

    const float* sb = src + (size_t)b * NPTS * 3;
    const float* tb = tgt + (size_t)b * NPTS * 3;
    const float* wb = wts + (size_t)b * NPTS;

    // ISA 16x4 f32 A layout: lane q=lane&15 holds row M=q; half=lane>>4 picks
    // K pair {0,1} (VGPR0/1, lanes 0-15) or {2,3} (lanes 16-31).
    // We tile M and N as 4 groups of 4 components; group g covers points
    // base + 4g + {0..3}; component index d = q&3 (d==3 -> homogeneous 1).
    const int halfid = lane >> 4;
    const int q      = lane & 15;
    const int g      = q >> 2;
    const int d      = q & 3;

    v8f c = {};                        // 16x16 f32 accumulator (8 VGPRs)
    const int base0 = wave * PPW;

    #pragma unroll 4
    for (int it = 0; it < PPW; it += 16) {
        const int p0 = base0 + it + 4 * g + 2 * halfid;
        const int p1 = p0 + 1;

        float w0 = wb[p0]; w0 = (w0 < WP_THRESH) ? 0.0f : w0;
        float w1 = wb[p1]; w1 = (w1 < WP_THRESH) ? 0.0f : w1;

        float s0 = 1.0f, s1 = 1.0f, t0 = 1.0f, t1 = 1.0f;
        if (d < 3) {
            s0 = sb[p0 * 3 + d];
            s1 = sb[p1 * 3 + d];
            t0 = tb[p0 * 3 + d];
            t1 = tb[p1 * 3 + d];
        }

        v2f av; av.x = s0;      av.y = s1;        // A: src components
        v2f bv; bv.x = w0 * t0; bv.y = w1 * t1;   // B: w * tgt components

        // D = A(16x4) * B(4x16) + C ; f32 throughout, K=4 point reduction.
        c = __builtin_amdgcn_wmma_f32_16x16x4_f32(
                false, av, false, bv, (short)0, c, false, false);
    }

    // Harvest diagonal 4x4 blocks of C into the shared 4x4 Gram matrix.
    // C layout: lanes 0-15 hold cols N=lane, rows M=r (VGPR r); lanes 16-31
    // hold cols N=lane-16, rows M=8+r.
    {
        const int Nc = lane & 15;
        const int gN = Nc >> 2;
        const int e  = Nc & 3;
        const int rs = halfid ? (gN - 2) * 4 : gN * 4;   // row-VGPR start of diag block
        if (rs == 0) {
            atomicAdd(&Hl[0 * 4 + e], c[0]);
            atomicAdd(&Hl[1 * 4 + e], c[1]);
            atomicAdd(&Hl[2 * 4 + e], c[2]);
            atomicAdd(&Hl[3 * 4 + e], c[3]);
        } else if (rs == 4) {
            atomicAdd(&Hl[0 * 4 + e], c[4]);
            atomicAdd(&Hl[1 * 4 + e], c[5]);
            atomicAdd(&Hl[2 * 4 + e], c[6]);
            atomicAdd(&Hl[3 * 4 + e], c[7]);
        }
    }
    __syncthreads();

    if (tid != 0) return;

    // ---------------- scalar epilogue: Kabsch from the 4x4 Gram -----------
    const float W   = Hl[15];
    const float inv = 1.0f / (W + WP_EPS);
    const float sw  = W * inv;

    float mus[3], mut[3];
    for (int i = 0; i < 3; ++i) { mus[i] = Hl[i * 4 + 3] * inv; mut[i] = Hl[12 + i] * inv; }

    // H = M' - (2 - sw) * mus * mut^T  (exact centered weighted covariance)
    float H3[3][3];
    const float cf = 2.0f - sw;
    for (int i = 0; i < 3; ++i)
        for (int j = 0; j < 3; ++j)
            H3[i][j] = Hl[i * 4 + j] * inv - cf * mus[i] * mut[j];

    // A = H^T H (symmetric) ; eigenvectors = right singular vectors V.
    float A[3][3];
    for (int i = 0; i < 3; ++i)
        for (int j = 0; j < 3; ++j) {
            float acc = 0.0f;
            for (int k = 0; k < 3; ++k) acc += H3[k][i] * H3[k][j];
            A[i][j] = acc;
        }

    float V[3][3] = {{1,0,0},{0,1,0},{0,0,1}};
    const int PQ[3][2] = {{0,1},{0,2},{1,2}};
    for (int sweep = 0; sweep < 10; ++sweep) {
        for (int r = 0; r < 3; ++r) {
            const int p = PQ[r][0], qq = PQ[r][1];
            const float apq = A[p][qq];
            if (fabsf(apq) < 1e-20f) continue;
            const float tau = (A[qq][qq] - A[p][p]) / (2.0f * apq);
            const float tt  = (tau >= 0.0f ? 1.0f : -1.0f) /
                              (fabsf(tau) + sqrtf(1.0f + tau * tau));
            const float cth = 1.0f / sqrtf(1.0f + tt * tt);
            const float sth = tt * cth;
            for (int k = 0; k < 3; ++k) {           // A = A * J
                const float akp = A[k][p], akq = A[k][qq];
                A[k][p]  = cth * akp - sth * akq;
                A[k][qq] = sth * akp + cth * akq;
            }
            for (int k = 0; k < 3; ++k) {           // A = J^T * A
                const float apk = A[p][k], aqk = A[qq][k];
                A[p][k]  = cth * apk - sth * aqk;
                A[qq][k] = sth * apk + cth * aqk;
            }
            for (int k = 0; k < 3; ++k) {           // V = V * J
                const float vkp = V[k][p], vkq = V[k][qq];
                V[k][p]  = cth * vkp - sth * vkq;
                V[k][qq] = sth * vkp + cth * vkq;
            }
        }
    }

    // Sort eigenpairs descending (columns of V).
    float lam[3] = {A[0][0], A[1][1], A[2][2]};
    for (int i = 0; i < 2; ++i)
        for (int j = i + 1; j < 3; ++j)
            if (lam[j] > lam[i]) {
                float tl = lam[i]; lam[i] = lam[j]; lam[j] = tl;
                for (int k = 0; k < 3; ++k) {
                    float tv = V[k][i]; V[k][i] = V[k][j]; V[k][j] = tv;
                }
            }

    float v1[3] = {V[0][0], V[1][0], V[2][0]};
    float v2[3] = {V[0][1], V[1][1], V[2][1]};

    float u1[3], u2[3], u3[3], v3[3];
    for (int i = 0; i < 3; ++i)
        u1[i] = H3[i][0] * v1[0] + H3[i][1] * v1[1] + H3[i][2] * v1[2];
    {
        float n = sqrtf(u1[0]*u1[0] + u1[1]*u1[1] + u1[2]*u1[2]);
        n = fmaxf(n, 1e-20f);
        for (int i = 0; i < 3; ++i) u1[i] /= n;
    }
    for (int i = 0; i < 3; ++i)
        u2[i] = H3[i][0] * v2[0] + H3[i][1] * v2[1] + H3[i][2] * v2[2];
    {
        const float dp = u2[0]*u1[0] + u2[1]*u1[1] + u2[2]*u1[2];
        for (int i = 0; i < 3; ++i) u2[i] -= dp * u1[i];
        float n = sqrtf(u2[0]*u2[0] + u2[1]*u2[1] + u2[2]*u2[2]);
        n = fmaxf(n, 1e-20f);
        for (int i = 0; i < 3; ++i) u2[i] /= n;
    }
    u3[0] = u1[1]*u2[2] - u1[2]*u2[1];
    u3[1] = u1[2]*u2[0] - u1[0]*u2[2];
    u3[2] = u1[0]*u2[1] - u1[1]*u2[0];
    v3[0] = v1[1]*v2[2] - v1[2]*v2[1];
    v3[1] = v1[2]*v2[0] - v1[0]*v2[2];
    v3[2] = v1[0]*v2[1] - v1[1]*v2[0];
    // u3 = u1 x u2 and v3 = v1 x v2 absorb Kabsch's det-sign fix exactly:
    // R = v1 u1^T + v2 u2^T + v3 u3^T == V diag(1,1,s) U^T.

    float R[3][3];
    for (int i = 0; i < 3; ++i)
        for (int j = 0; j < 3; ++j)
            R[i][j] = v1[i]*u1[j] + v2[i]*u2[j] + v3[i]*u3[j];

    float* Rout = out + (size_t)b * 9;
    float* Tout = out + (size_t)B * 9 + (size_t)b * 3;
    for (int i = 0; i < 3; ++i) {
        for (int j = 0; j < 3; ++j) Rout[i * 3 + j] = R[i][j];
        Tout[i] = mut[i] - (R[i][0]*mus[0] + R[i][1]*mus[1] + R[i][2]*mus[2]);
    }
}

extern "C" void kernel_launch(void* const* d_in, const int* in_sizes, int n_in,
                              void* d_out, int out_size, void* d_ws, size_t ws_size,
                              hipStream_t stream) {
    const float* src = (const float*)d_in[0];   // (B, N, 3) f32
    const float* tgt = (const float*)d_in[1];   // (B, N, 3) f32
    const float* wts = (const float*)d_in[2];   // (B, N)    f32
    float* out = (float*)d_out;                 // R (B*9) then t (B*3)

    const int B = in_sizes[2] / NPTS;           // reference: 4096 x 2048

    wproc_kernel<<<B, BLKT, 0, stream>>>(src, tgt, wts, out, B);
}